// Linear_46265387713121
// MI455X (gfx1250) — compile-verified
//
#include <hip/hip_runtime.h>

typedef __attribute__((ext_vector_type(2))) float v2f;
typedef __attribute__((ext_vector_type(8))) float v8f;

#define IBP_IN 8192
#define IBP_OUT 8192
#define WAVES_PER_WG 16   // 512 threads, wave32
#define K_SLICE (IBP_IN / WAVES_PER_WG)   // 512 rows per wave

// Kernel 1: c = (l+u)/2 into ws[0..8191], r = (u-l)/2 into ws[8192..16383].
__global__ void ibp_cr_precompute(const float* __restrict__ l,
                                  const float* __restrict__ u,
                                  float* __restrict__ cr) {
  int i = blockIdx.x * blockDim.x + threadIdx.x;
  if (i < IBP_IN) {
    float lv = l[i], uv = u[i];
    cr[i]          = 0.5f * (lv + uv);   // c
    cr[IBP_IN + i] = 0.5f * (uv - lv);   // r
  }
}

// Kernel 2: per workgroup, one 16-column output tile.
// Each wave handles a K-slice; WMMA f32 16x16x4 accumulates
//   row 0 of D: sum_k c[k] * W[k][j]      (A row 0 = c chunk, lanes 0/16)
//   row 1 of D: sum_k r[k] * |W[k][j]|    (A row 1 = r chunk, lanes 1/17)
// bound_l = row0 - row1 + bias; bound_u = row0 + row1 + bias.
// Weights streamed once non-temporally in GVS (saddr + 32-bit voffset) form.
// c/r come from ONE per-lane load: r-lanes read at a +8192-element offset in
// the packed cr array; offset tracked as a 32-bit byte offset so this load
// also lowers to saddr + voffset + immediate.
__global__ __launch_bounds__(WAVES_PER_WG * 32)
void ibp_gemv_wmma(const float* __restrict__ weight,
                   const float* __restrict__ bias,
                   const float* __restrict__ cr,
                   float* __restrict__ out) {
  __shared__ float red[WAVES_PER_WG * 32];

  const int lane = threadIdx.x & 31;
  const int wave = threadIdx.x >> 5;
  const int col  = lane & 15;           // N index (output column within tile)
  const int half = (lane >> 4) & 1;     // lanes 16-31 hold K+2 / K+3
  const int j0   = blockIdx.x * 16;

  const int kBegin = wave * K_SLICE;

  // B layout (4x16 f32): VGPR0 = {K=0 | K=2}, VGPR1 = {K=1 | K=3}, lanes = N.
  // Lane loads W[k + 2*half + {0,1}][j0 + col].
  // 32-bit element offsets: weight is 64M elements (256 MB) < 2^31 bytes.
  uint32_t wIdx = (uint32_t)(kBegin + 2 * half) * IBP_OUT +
                  (uint32_t)(j0 + col);

  // Per-lane byte offset into packed {c | r} array: r-lanes (col==1) read the
  // r half.  All offsets stay 8-byte aligned.
  const char* xB = (const char*)cr;
  uint32_t xOff = ((uint32_t)(kBegin + 2 * half) +
                   ((col == 1) ? (uint32_t)IBP_IN : 0u)) * 4u;

  v8f acc = {};

#pragma unroll 8
  for (int it = 0; it < K_SLICE / 4; ++it) {
    float w0 = __builtin_nontemporal_load(weight + wIdx);
    float w1 = __builtin_nontemporal_load(weight + wIdx + IBP_OUT);
    v2f xv = *(const v2f*)(xB + xOff);   // c-pair for c-lanes, r-pair for r-lanes

    // A layout (16x4 f32): lane m<16 holds A[m][0],A[m][1]; lane m>=16 holds
    // A[m-16][2],A[m-16][3].  Row 0 = c (lanes 0,16), row 1 = r (lanes 1,17).
    v2f a_c, a_r, b_w, b_a;
    a_c.x = (col == 0) ? xv.x : 0.0f;
    a_c.y = (col == 0) ? xv.y : 0.0f;
    a_r.x = (col == 1) ? xv.x : 0.0f;
    a_r.y = (col == 1) ? xv.y : 0.0f;
    b_w.x = w0;
    b_w.y = w1;
    b_a.x = __builtin_fabsf(w0);
    b_a.y = __builtin_fabsf(w1);

    // acc row0 += c . W ; acc row1 += r . |W|   (accumulator chaining, D->C)
    acc = __builtin_amdgcn_wmma_f32_16x16x4_f32(false, a_c, false, b_w,
                                                (short)0, acc, false, false);
    acc = __builtin_amdgcn_wmma_f32_16x16x4_f32(false, a_r, false, b_a,
                                                (short)0, acc, false, false);

    wIdx += 4u * IBP_OUT;
    xOff += 16u;
  }

  // D layout: M=0 row -> acc[0] lanes 0-15; M=1 row -> acc[1] lanes 0-15.
  if (lane < 16) {
    red[wave * 32 + lane]      = acc[0];  // partial y_c for column j0+lane
    red[wave * 32 + 16 + lane] = acc[1];  // partial y_r
  }
  __syncthreads();

  if (threadIdx.x < 16) {
    int t = threadIdx.x;
    float yc = 0.0f, yr = 0.0f;
#pragma unroll
    for (int w = 0; w < WAVES_PER_WG; ++w) {
      yc += red[w * 32 + t];
      yr += red[w * 32 + 16 + t];
    }
    float b = bias[j0 + t];
    out[j0 + t]           = yc - yr + b;  // bound_l
    out[IBP_OUT + j0 + t] = yc + yr + b;  // bound_u
  }
}

extern "C" void kernel_launch(void* const* d_in, const int* in_sizes, int n_in,
                              void* d_out, int out_size, void* d_ws, size_t ws_size,
                              hipStream_t stream) {
  const float* l      = (const float*)d_in[0];
  const float* u      = (const float*)d_in[1];
  const float* weight = (const float*)d_in[2];
  const float* bias   = (const float*)d_in[3];
  float* out = (float*)d_out;

  float* crArr = (float*)d_ws;         // 16384 floats: c then r (64 KB)

  ibp_cr_precompute<<<IBP_IN / 256, 256, 0, stream>>>(l, u, crArr);
  ibp_gemv_wmma<<<IBP_OUT / 16, WAVES_PER_WG * 32, 0, stream>>>(weight, bias,
                                                                crArr, out);
}